// FluxPosRegressor_43482248904948
// MI455X (gfx1250) — compile-verified
//
#include <hip/hip_runtime.h>

// Problem constants (reference: N=64, H=1024, W=1024, f32 in/out).
#define NF  64
#define HH  1024
#define WW  1024
#define PP  (2 * HH - 2)   // mirror period (scipy mode='mirror'), 2046
#define RPB 8              // output rows per block
#define SROWS (RPB + 1)    // staged input rows per block

// Flip to 0 to fall back to VGPR-staged LDS fill (no async engine).
#define USE_ASYNC_LDS 1

typedef __attribute__((address_space(3))) float lds_float;

__device__ __forceinline__ int mirror_idx(int i) {
  // scipy/jax mode='mirror': reflect about edge centers, period 2*(H-1).
  if ((unsigned)i < (unsigned)HH) return i;   // fast path, almost always taken
  int m = i % PP;
  if (m < 0) m += PP;
  return (m >= HH) ? (PP - m) : m;
}

__global__ __launch_bounds__(256)
void FluxPosRegressor_43482248904948_kernel(const float* __restrict__ in,
                                            const float* __restrict__ flux,
                                            const float* __restrict__ noise,
                                            const float* __restrict__ dydx,
                                            float* __restrict__ out) {
  __shared__ __align__(16) float rows[SROWS * WW];   // 36 KB of the 320 KB WGP LDS

  const int t  = threadIdx.x;        // 0..255
  const int y0 = blockIdx.x * RPB;   // first output row of this block
  const int n  = blockIdx.y;         // frame

  // Per-frame shift decomposition: floor(y - dy) = y + floor(-dy) for integer y.
  const float dy  = dydx[2 * n + 0];
  const float dx  = dydx[2 * n + 1];
  const float oyf = floorf(-dy);
  const float oxf = floorf(-dx);
  const float fy  = -dy - oyf;       // fractional weights, constant per frame
  const float fx  = -dx - oxf;
  const int   oy  = (int)oyf;
  const int   ox  = (int)oxf;

  const size_t frame = (size_t)n * HH * WW;

  // ---- Stage SROWS mirrored input rows into LDS (async global->LDS DMA) ----
  // Each thread moves 16B (one float4) of each staged row: 256 thr * 16B = 4 KB = one row.
#if USE_ASYNC_LDS
  lds_float* lbase = (lds_float*)rows;
#endif
#pragma unroll
  for (int j = 0; j < SROWS; ++j) {
    const int r = mirror_idx(y0 + oy + j);
    const float* src = in + frame + (size_t)r * WW + 4 * t;
#if USE_ASYNC_LDS
    unsigned int       ldsoff = (unsigned int)(uintptr_t)(lbase + j * WW + 4 * t);
    unsigned long long ga     = (unsigned long long)(uintptr_t)src;
    asm volatile("global_load_async_to_lds_b128 %0, %1, off"
                 :: "v"(ldsoff), "v"(ga)
                 : "memory");
#else
    *(float4*)&rows[j * WW + 4 * t] = *(const float4*)src;
#endif
  }
#if USE_ASYNC_LDS
  asm volatile("s_wait_asynccnt 0x0" ::: "memory");
#endif
  __syncthreads();

  // ---- Bilinear combine: 4 taps with per-frame constant weights ----
  const float w11 = fy * fx;
  const float w10 = fy - w11;          // fy*(1-fx)
  const float w01 = fx - w11;          // (1-fy)*fx
  const float w00 = 1.0f - fy - w01;   // (1-fy)*(1-fx)
  const float fl  = flux[n];
  const float no  = noise[n];

  // Mirrored column indices: computed once, reused for all 8 output rows.
  int c[5];
#pragma unroll
  for (int k = 0; k < 5; ++k) c[k] = mirror_idx(4 * t + ox + k);

#pragma unroll
  for (int i = 0; i < RPB; ++i) {
    const float* ra = &rows[i * WW];         // mirror(y+oy)   for y = y0+i
    const float* rb = &rows[(i + 1) * WW];   // mirror(y+oy+1)
    float4 res;
    float* rp = (float*)&res;
#pragma unroll
    for (int k = 0; k < 4; ++k) {
      const float v00 = ra[c[k]];
      const float v01 = ra[c[k + 1]];
      const float v10 = rb[c[k]];
      const float v11 = rb[c[k + 1]];
      rp[k] = (w00 * v00 + w01 * v01 + w10 * v10 + w11 * v11) * fl + no;
    }
    *(float4*)(out + frame + (size_t)(y0 + i) * WW + 4 * t) = res;  // aligned b128
  }
}

extern "C" void kernel_launch(void* const* d_in, const int* in_sizes, int n_in,
                              void* d_out, int out_size, void* d_ws, size_t ws_size,
                              hipStream_t stream) {
  (void)in_sizes; (void)n_in; (void)out_size; (void)d_ws; (void)ws_size;
  const float* in    = (const float*)d_in[0];   // [1,N,H,W,1] f32
  const float* flux  = (const float*)d_in[1];   // [N,1,1,1]   f32
  const float* noise = (const float*)d_in[2];   // [N,1,1,1]   f32
  const float* dydx  = (const float*)d_in[3];   // [N,2]       f32
  float* out = (float*)d_out;

  dim3 grid(HH / RPB, NF);   // 128 x 64 blocks
  dim3 block(256);           // 8 wave32s
  FluxPosRegressor_43482248904948_kernel<<<grid, block, 0, stream>>>(in, flux, noise, dydx, out);
}